// torch_Gent_56994216018357
// MI455X (gfx1250) — compile-verified
//
#include <hip/hip_runtime.h>

// ---------------------------------------------------------------------------
// Gent hyperelastic energy, per-element over N 3x3 deformation gradients.
// Memory-bound: 40 B/element, 160 MB total -> ~6.9 us at 23.3 TB/s HBM.
// Strategy: coalesced b128 staging of the 36-byte AoS records through LDS
// via the CDNA5 async global->LDS DMA path (ASYNCcnt-tracked), register
// compute, non-temporal b128 result stores. Interior blocks take a fully
// unguarded fast path (no exec-mask churn); only the last block does
// bounds-checked staging/stores.
// ---------------------------------------------------------------------------

#ifndef __has_builtin
#define __has_builtin(x) 0
#endif

#if __has_builtin(__builtin_amdgcn_global_load_async_to_lds_b128) && \
    __has_builtin(__builtin_amdgcn_s_wait_asynccnt)
#define GENT_ASYNC_LDS 1
#else
#define GENT_ASYNC_LDS 0
#endif

typedef float v4f __attribute__((ext_vector_type(4)));
typedef int   v4i __attribute__((ext_vector_type(4)));

#define AS1 __attribute__((address_space(1)))
#define AS3 __attribute__((address_space(3)))
// Builtin prototype (probe-confirmed): (AS1 int4*, AS3 int4*, imm off, imm cpol).
// Generic->LDS: low 32 bits of a generic LDS pointer are the LDS byte offset
// (aperture encoding, ISA 10.2); compiler folds the round-trip to addrspacecast.
#define TO_AS1_V4I(p) ((AS1 v4i*)(unsigned long long)(p))
#define TO_AS3_V4I(p) ((AS3 v4i*)(unsigned int)(unsigned long long)(p))

#if GENT_ASYNC_LDS
#define STAGE_V4(k_, g4_)                                            \
    __builtin_amdgcn_global_load_async_to_lds_b128(                  \
        TO_AS1_V4I(x + (size_t)(g4_) * 4),                           \
        TO_AS3_V4I(&tile4[(k_) * kBlock + t]), 0, 0)
#else
#define STAGE_V4(k_, g4_) \
    (tile4[(k_) * kBlock + t] = reinterpret_cast<const v4f*>(x)[(g4_)])
#endif

namespace {
constexpr int kBlock         = 256;
constexpr int kEPT           = 4;                       // elements per thread
constexpr int kElemsPerBlock = kBlock * kEPT;           // 1024
constexpr int kVec4PerBlock  = kElemsPerBlock * 9 / 4;  // 2304 float4 per block
constexpr int kV4PerThread   = kVec4PerBlock / kBlock;  // 9

constexpr float kG   = 75000.0f / (2.0f * (1.0f + 0.3f));          // shear modulus
constexpr float kKm  = 75000.0f / (3.0f * (1.0f - 2.0f * 0.3f));   // bulk modulus
constexpr float kJm  = 100.0f;
constexpr float kLn2 = 0.69314718055994530942f;
}  // namespace

__global__ __launch_bounds__(kBlock)
void gent_energy_kernel(const float* __restrict__ x, float* __restrict__ w, int n)
{
    __shared__ v4f tile4[kVec4PerBlock];  // 36 KB: 1024 elements x 9 floats
    float* tileF = reinterpret_cast<float*>(tile4);

    const int t        = threadIdx.x;
    const int elemBase = blockIdx.x * kElemsPerBlock;
    const bool full    = (elemBase + kElemsPerBlock) <= n;  // interior block?

    // ---- Stage this block's 9216 floats into LDS, flat & coalesced (b128) ----
    if (full) {
        // Fast path: 9 back-to-back unguarded async b128 transfers, EXEC all-1s.
        const int base4 = (elemBase * 9) >> 2;
        #pragma unroll
        for (int k = 0; k < kV4PerThread; ++k)
            STAGE_V4(k, base4 + k * kBlock + t);
    } else {
        // Last block only: per-lane bounds-checked staging + ragged tail.
        const int flatN  = n * 9;
        const int flatN4 = flatN >> 2;
        const int base4  = (elemBase * 9) >> 2;
        #pragma unroll
        for (int k = 0; k < kV4PerThread; ++k) {
            const int g4 = base4 + k * kBlock + t;
            if (g4 < flatN4)
                STAGE_V4(k, g4);
        }
        const int fi    = (flatN4 << 2) + t;   // up to 3 ragged floats
        const int local = fi - elemBase * 9;
        if (fi < flatN && local >= 0 && local < kElemsPerBlock * 9)
            tileF[local] = x[fi];
    }
#if GENT_ASYNC_LDS
    __builtin_amdgcn_s_wait_asynccnt(0);
#endif
    __syncthreads();

    // ---- Each thread pulls its 4 elements (36 contiguous floats) from LDS.
    //      Lane L's 9x ds_load_b128 at byte 144*L: lanes 0-15 tile all 64
    //      banks, lanes 16-31 repeat -> 2 cycles/op, the b128 floor. ----
    v4f vreg[9];
    #pragma unroll
    for (int q = 0; q < 9; ++q)
        vreg[q] = tile4[t * 9 + q];
    const float* f36 = reinterpret_cast<const float*>(vreg);

    float out[kEPT];
    #pragma unroll
    for (int e = 0; e < kEPT; ++e) {
        const float* f = f36 + e * 9;

        // I1 = tr(F^T F) = sum of squares of all 9 entries
        float I1 = f[0] * f[0];
        #pragma unroll
        for (int j = 1; j < 9; ++j) I1 = fmaf(f[j], f[j], I1);

        // J = det(F), cofactor expansion along row 0
        const float c0 = f[4] * f[8] - f[5] * f[7];
        const float c1 = f[3] * f[8] - f[5] * f[6];
        const float c2 = f[3] * f[7] - f[4] * f[6];
        const float J  = f[0] * c0 - f[1] * c1 + f[2] * c2;

        const float l2J   = log2f(J);
        const float lnJ   = l2J * kLn2;
        const float Jm23  = exp2f((-2.0f / 3.0f) * l2J);  // J^(-2/3)
        const float I1bar = I1 * Jm23;

        const float vol = 0.5f * (J * J - 1.0f) - lnJ;
        const float arg = 1.0f - (I1bar - 3.0f) * (1.0f / kJm);
        const float lnA = log2f(arg) * kLn2;
        const float v2  = vol * vol;

        out[e] = (-0.5f * kG * kJm) * lnA + (0.5f * kKm) * (v2 * v2);
    }

    // ---- Coalesced 16-B non-temporal stores (keep input L2-resident) ----
    const int i0 = elemBase + t * kEPT;
    if (full) {
        v4f o = {out[0], out[1], out[2], out[3]};
        __builtin_nontemporal_store(o, reinterpret_cast<v4f*>(w + i0));
    } else {
        #pragma unroll
        for (int e = 0; e < kEPT; ++e)
            if (i0 + e < n) w[i0 + e] = out[e];
    }
}

extern "C" void kernel_launch(void* const* d_in, const int* in_sizes, int n_in,
                              void* d_out, int out_size, void* d_ws, size_t ws_size,
                              hipStream_t stream)
{
    (void)n_in; (void)out_size; (void)d_ws; (void)ws_size;
    const float* x = reinterpret_cast<const float*>(d_in[0]);
    float*       w = reinterpret_cast<float*>(d_out);
    const int n      = in_sizes[0] / 9;
    const int blocks = (n + kElemsPerBlock - 1) / kElemsPerBlock;
    if (blocks > 0)
        gent_energy_kernel<<<blocks, kBlock, 0, stream>>>(x, w, n);
}